// DE_35502199668823
// MI455X (gfx1250) — compile-verified
//
#include <hip/hip_runtime.h>
#include <hip/hip_bf16.h>

#define TPB     256
#define BPB     32          // blocks per batch
#define NBATCH  64
#define NBINS   256
#define NWAVE   (TPB / 32)  // wave32 on gfx1250
#define NBUF    4           // async staging depth (power of 2)

// Monotone float -> uint key (order-preserving for all floats)
__device__ __forceinline__ unsigned fkey(float x) {
    unsigned u = __float_as_uint(x);
    return (u & 0x80000000u) ? ~u : (u | 0x80000000u);
}
__device__ __forceinline__ float kinv(unsigned k) {
    unsigned u = (k & 0x80000000u) ? (k & 0x7FFFFFFFu) : ~k;
    return __uint_as_float(u);
}

// ---------------- Pass 1: per-batch min/max ----------------
__global__ void DE_minmax_kernel(const float* __restrict__ x,
                                 unsigned* __restrict__ mink,
                                 unsigned* __restrict__ maxk,
                                 int n_per_batch) {
    const int b     = blockIdx.y;
    const int chunk = (n_per_batch + BPB - 1) / BPB;
    const int count = min(chunk, n_per_batch - (int)blockIdx.x * chunk);
    const float* p  = x + (long long)b * n_per_batch + (long long)blockIdx.x * chunk;

    float lmin = __int_as_float(0x7F800000);   // +inf
    float lmax = -lmin;

    const int nv4 = count >> 2;
    const float4* p4 = (const float4*)p;
    for (int i = threadIdx.x; i < nv4; i += TPB) {
        // CDNA5 global_prefetch_b8: pull future cachelines toward L2/WGP$
        __builtin_prefetch(p4 + i + 8 * TPB, 0, 0);
        float4 v = p4[i];
        lmin = fminf(lmin, fminf(fminf(v.x, v.y), fminf(v.z, v.w)));
        lmax = fmaxf(lmax, fmaxf(fmaxf(v.x, v.y), fmaxf(v.z, v.w)));
    }
    for (int i = (nv4 << 2) + threadIdx.x; i < count; i += TPB) {
        float v = p[i];
        lmin = fminf(lmin, v);
        lmax = fmaxf(lmax, v);
    }

    __shared__ float smin[TPB], smax[TPB];
    smin[threadIdx.x] = lmin;
    smax[threadIdx.x] = lmax;
    __syncthreads();
    for (int s = TPB / 2; s > 0; s >>= 1) {
        if (threadIdx.x < s) {
            smin[threadIdx.x] = fminf(smin[threadIdx.x], smin[threadIdx.x + s]);
            smax[threadIdx.x] = fmaxf(smax[threadIdx.x], smax[threadIdx.x + s]);
        }
        __syncthreads();
    }
    if (threadIdx.x == 0) {
        atomicMin(&mink[b], fkey(smin[0]));
        atomicMax(&maxk[b], fkey(smax[0]));
    }
}

// ---------------- Pass 2: histogram with async global->LDS staging ----------------
__global__ void DE_hist_kernel(const float* __restrict__ x,
                               const unsigned* __restrict__ mink,
                               const unsigned* __restrict__ maxk,
                               unsigned* __restrict__ ghist,
                               int n_per_batch) {
    const int b = blockIdx.y;
    const float mn = kinv(mink[b]);
    const float mx = kinv(maxk[b]);
    // torch.histc fixup: if min == max, use [min-1, max+1]
    const float rng   = (mx > mn) ? (mx - mn) : 2.0f;
    const float lo    = (mx > mn) ? mn : (mn - 1.0f);
    const float scale = (float)NBINS / rng;   // matches reference: NBINS/rng first

    __shared__ unsigned sh[NWAVE * NBINS];    // 8 KB per-wave private histograms
    __shared__ float4   stage[NBUF][TPB];     // 16 KB async staging buffers
    for (int i = threadIdx.x; i < NWAVE * NBINS; i += TPB) sh[i] = 0u;
    __syncthreads();
    unsigned* myh = sh + (threadIdx.x >> 5) * NBINS;

    const int chunk = (n_per_batch + BPB - 1) / BPB;
    const int count = min(chunk, n_per_batch - (int)blockIdx.x * chunk);
    const float* p  = x + (long long)b * n_per_batch + (long long)blockIdx.x * chunk;

    const int nv4 = count >> 2;
    const float4* p4 = (const float4*)p;
    const int tid = threadIdx.x;
    const int T = nv4 / TPB;                  // full tiles of TPB float4s

    // LDS byte offset of this thread's slot in buffer 0 (low 32 bits of the
    // generic LDS address == workgroup-relative LDS offset on amdgcn)
    const unsigned lds_base = (unsigned)(uintptr_t)(&stage[0][tid]);

    // Prologue: fill the pipeline
    const int prologue = (T < NBUF) ? T : NBUF;
    for (int k = 0; k < prologue; ++k) {
        const float4* g = p4 + (size_t)k * TPB + tid;
        unsigned l = lds_base + (unsigned)(k * TPB * sizeof(float4));
        asm volatile("global_load_async_to_lds_b128 %0, %1, off"
                     :: "v"(l), "v"((unsigned long long)(uintptr_t)g)
                     : "memory");
    }

    for (int t = 0; t < T; ++t) {
        // Wait until tile t has landed: allow min(NBUF-1, tiles-left) pending.
        const int rem  = T - 1 - t;
        const int pend = (rem < NBUF - 1) ? rem : (NBUF - 1);
        switch (pend) {
            case 3:  asm volatile("s_wait_asynccnt 0x3" ::: "memory"); break;
            case 2:  asm volatile("s_wait_asynccnt 0x2" ::: "memory"); break;
            case 1:  asm volatile("s_wait_asynccnt 0x1" ::: "memory"); break;
            default: asm volatile("s_wait_asynccnt 0x0" ::: "memory"); break;
        }
        float4 v = stage[t & (NBUF - 1)][tid];  // ds_load_b128
        int i0 = min(max((int)((v.x - lo) * scale), 0), NBINS - 1);
        int i1 = min(max((int)((v.y - lo) * scale), 0), NBINS - 1);
        int i2 = min(max((int)((v.z - lo) * scale), 0), NBINS - 1);
        int i3 = min(max((int)((v.w - lo) * scale), 0), NBINS - 1);
        atomicAdd(&myh[i0], 1u);   // ds_add_u32
        atomicAdd(&myh[i1], 1u);
        atomicAdd(&myh[i2], 1u);
        atomicAdd(&myh[i3], 1u);
        // Refill the slot we just consumed
        if (t + NBUF < T) {
            const float4* g = p4 + (size_t)(t + NBUF) * TPB + tid;
            unsigned l = lds_base + (unsigned)((t & (NBUF - 1)) * TPB * sizeof(float4));
            asm volatile("global_load_async_to_lds_b128 %0, %1, off"
                         :: "v"(l), "v"((unsigned long long)(uintptr_t)g)
                         : "memory");
        }
    }

    // Tail float4s not covered by full tiles (direct loads)
    for (int i = T * TPB + tid; i < nv4; i += TPB) {
        float4 v = p4[i];
        int i0 = min(max((int)((v.x - lo) * scale), 0), NBINS - 1);
        int i1 = min(max((int)((v.y - lo) * scale), 0), NBINS - 1);
        int i2 = min(max((int)((v.z - lo) * scale), 0), NBINS - 1);
        int i3 = min(max((int)((v.w - lo) * scale), 0), NBINS - 1);
        atomicAdd(&myh[i0], 1u);
        atomicAdd(&myh[i1], 1u);
        atomicAdd(&myh[i2], 1u);
        atomicAdd(&myh[i3], 1u);
    }
    // Scalar tail
    for (int i = (nv4 << 2) + tid; i < count; i += TPB) {
        int i0 = min(max((int)((p[i] - lo) * scale), 0), NBINS - 1);
        atomicAdd(&myh[i0], 1u);
    }
    __syncthreads();

    // Reduce per-wave histograms, one global atomic per bin per block
    for (int t = tid; t < NBINS; t += TPB) {
        unsigned s = 0u;
        for (int w = 0; w < NWAVE; ++w) s += sh[w * NBINS + t];
        if (s) atomicAdd(&ghist[b * NBINS + t], s);
    }
}

// ---------------- Pass 3: entropy per batch, mean over batches ----------------
__global__ void DE_entropy_kernel(const unsigned* __restrict__ ghist,
                                  float* __restrict__ out) {
    __shared__ float red[NBINS];
    const int t = threadIdx.x;
    float acc = 0.0f;
    for (int b = 0; b < NBATCH; ++b) {
        float c = (float)ghist[b * NBINS + t];
        red[t] = c;
        __syncthreads();
        for (int s = NBINS / 2; s > 0; s >>= 1) {
            if (t < s) red[t] += red[t + s];
            __syncthreads();
        }
        float total = red[0];
        __syncthreads();
        float pr = c / total;
        float term = (c > 0.0f) ? (-pr * log2f(pr)) : 0.0f;
        red[t] = term;
        __syncthreads();
        for (int s = NBINS / 2; s > 0; s >>= 1) {
            if (t < s) red[t] += red[t + s];
            __syncthreads();
        }
        if (t == 0) acc += red[0];
        __syncthreads();
    }
    if (t == 0) out[0] = acc / (float)NBATCH;
}

extern "C" void kernel_launch(void* const* d_in, const int* in_sizes, int n_in,
                              void* d_out, int out_size, void* d_ws, size_t ws_size,
                              hipStream_t stream) {
    const float* x = (const float*)d_in[0];
    float* out = (float*)d_out;

    const int total = in_sizes[0];
    const int n_per_batch = total / NBATCH;   // 3*512*512 = 786432

    // Workspace layout: [min keys (64)] [max keys (64)] [hist (64*256)]
    unsigned* mink  = (unsigned*)d_ws;
    unsigned* maxk  = mink + NBATCH;
    unsigned* ghist = maxk + NBATCH;

    hipMemsetAsync(mink, 0xFF, NBATCH * sizeof(unsigned), stream);            // +inf key
    hipMemsetAsync(maxk, 0x00, NBATCH * sizeof(unsigned), stream);            // -inf key
    hipMemsetAsync(ghist, 0x00, NBATCH * NBINS * sizeof(unsigned), stream);

    dim3 grid(BPB, NBATCH);
    DE_minmax_kernel<<<grid, TPB, 0, stream>>>(x, mink, maxk, n_per_batch);
    DE_hist_kernel<<<grid, TPB, 0, stream>>>(x, mink, maxk, ghist, n_per_batch);
    DE_entropy_kernel<<<1, NBINS, 0, stream>>>(ghist, out);
}